// GAT_1288490188987
// MI455X (gfx1250) — compile-verified
//
#include <hip/hip_runtime.h>
#include <hip/hip_bf16.h>

typedef __attribute__((ext_vector_type(2))) float v2f;
typedef __attribute__((ext_vector_type(8))) float v8f;

#define NNODES 50000
#define NEDGES 800000
#define WDIM   256     // HEADS*HID
#define NHEADS 8
#define NHID   32
#define NGRAPH 256

// ---------- helpers: monotone float<->uint encoding for atomic max ----------
__device__ __forceinline__ unsigned fenc(float x) {
    unsigned u = __float_as_uint(x);
    return (u & 0x80000000u) ? ~u : (u | 0x80000000u);
}
__device__ __forceinline__ float fdec(unsigned e) {
    unsigned u = (e & 0x80000000u) ? (e & 0x7FFFFFFFu) : ~e;
    return __uint_as_float(u);
}

// ---------- generic fill ----------
__global__ void fill_u32(unsigned* __restrict__ p, unsigned v, long n) {
    long i = (long)blockIdx.x * blockDim.x + threadIdx.x;
    if (i < n) p[i] = v;
}

// ---------- fp32 WMMA GEMM: C[M,N] = A[M,K] @ B[K,N] ----------
// One wave computes a 16(M) x 64(N) strip = 4 adjacent 16x16 tiles, so the
// A fragment (one global_load_b64 per K-step) feeds 4 v_wmma issues.
// Layout per CDNA5 ISA 7.12.2:
//   A 16x4 : lanes 0-15 hold M=lane, VGPR0=K0, VGPR1=K1; lanes 16-31 hold K2/K3
//   B 4x16 : VGPR0 lanes 0-15 row K0 (N=lane), lanes 16-31 row K2; VGPR1 K1/K3
//   C 16x16: VGPR r, lanes 0-15 -> M=r, lanes 16-31 -> M=r+8, N=lane&15
__global__ void gemm_f32_wmma(const float* __restrict__ A, const float* __restrict__ B,
                              float* __restrict__ C, int M, int N, int K) {
    int wave   = (blockIdx.x * blockDim.x + threadIdx.x) >> 5;
    int lane   = threadIdx.x & 31;
    int nstrip = N >> 6;                       // 64-wide N strips
    int m0 = (wave / nstrip) << 4;
    int n0 = (wave % nstrip) << 6;
    if (m0 >= M) return;
    int hi = lane >> 4;                        // half-wave: K offset 0 or 2
    int l  = lane & 15;

    v8f c0 = {}, c1 = {}, c2 = {}, c3 = {};
    const float* arow  = A + (size_t)(m0 + l) * K + (hi << 1);
    const float* bbase = B + (size_t)(hi << 1) * N + n0 + l;

    for (int k = 0; k < K; k += 4) {
        v2f a = *(const v2f*)(arow + k);       // K = k+2hi, k+2hi+1
        const float* bk0 = bbase + (size_t)k * N;   // row k+2hi
        const float* bk1 = bk0 + N;                 // row k+2hi+1
        v2f b0, b1, b2, b3;
        b0.x = bk0[0];   b0.y = bk1[0];
        b1.x = bk0[16];  b1.y = bk1[16];
        b2.x = bk0[32];  b2.y = bk1[32];
        b3.x = bk0[48];  b3.y = bk1[48];
        c0 = __builtin_amdgcn_wmma_f32_16x16x4_f32(false, a, false, b0, (short)0, c0, false, false);
        c1 = __builtin_amdgcn_wmma_f32_16x16x4_f32(false, a, false, b1, (short)0, c1, false, false);
        c2 = __builtin_amdgcn_wmma_f32_16x16x4_f32(false, a, false, b2, (short)0, c2, false, false);
        c3 = __builtin_amdgcn_wmma_f32_16x16x4_f32(false, a, false, b3, (short)0, c3, false, false);
    }
    float* crow = C + (size_t)(m0 + (hi << 3)) * N + n0 + l;
#pragma unroll
    for (int r = 0; r < 8; ++r) {
        crow[(size_t)r * N +  0] = c0[r];
        crow[(size_t)r * N + 16] = c1[r];
        crow[(size_t)r * N + 32] = c2[r];
        crow[(size_t)r * N + 48] = c3[r];
    }
}

// ---------- per-(node,head) attention pre-logits ----------
__global__ void compute_al(const float* __restrict__ h, const float* __restrict__ as_,
                           const float* __restrict__ ad_, float* __restrict__ als,
                           float* __restrict__ ald, int n_nodes) {
    int i = blockIdx.x * blockDim.x + threadIdx.x;     // node*8 + head
    if (i >= n_nodes * NHEADS) return;
    int head = i & 7;
    const float* hp = h + (size_t)(i >> 3) * WDIM + head * NHID;
    const float* sp = as_ + head * NHID;
    const float* dp = ad_ + head * NHID;
    float s = 0.f, d = 0.f;
#pragma unroll
    for (int c = 0; c < NHID; ++c) { float v = hp[c]; s += v * sp[c]; d += v * dp[c]; }
    als[i] = s; ald[i] = d;
}

// ---------- edge pass 1: leaky-relu logits + segment max (atomic on encoded uint) ----------
__global__ void edge_logits(const int* __restrict__ src, const int* __restrict__ dst,
                            const float* __restrict__ als, const float* __restrict__ ald,
                            float* __restrict__ ep, unsigned* __restrict__ menc) {
    long i = (long)blockIdx.x * blockDim.x + threadIdx.x;
    long tot = (long)(NEDGES + NNODES) * NHEADS;
    if (i >= tot) return;
    int e = (int)(i >> 3), head = (int)(i & 7);
    int s, d;
    if (e < NEDGES) { s = src[e]; d = dst[e]; } else { s = d = e - NEDGES; }
    float v = als[s * NHEADS + head] + ald[d * NHEADS + head];
    v = v > 0.f ? v : 0.2f * v;                        // leaky_relu slope 0.2
    ep[i] = v;
    atomicMax(&menc[d * NHEADS + head], fenc(v));
}

// ---------- edge pass 2: exp(e - m[dst]) + segment sum ----------
__global__ void edge_exp(const int* __restrict__ dst, float* __restrict__ ep,
                         const unsigned* __restrict__ menc, float* __restrict__ denom) {
    long i = (long)blockIdx.x * blockDim.x + threadIdx.x;
    long tot = (long)(NEDGES + NNODES) * NHEADS;
    if (i >= tot) return;
    int e = (int)(i >> 3), head = (int)(i & 7);
    int d = (e < NEDGES) ? dst[e] : (e - NEDGES);
    float m = fdec(menc[d * NHEADS + head]);
    float p = expf(ep[i] - m);
    ep[i] = p;
    atomicAdd(&denom[d * NHEADS + head], p);
}

// ---------- edge pass 3: out[dst] += alpha * h[src] (block = 256 channels of one edge) ----------
__global__ void edge_agg(const int* __restrict__ src, const int* __restrict__ dst,
                         const float* __restrict__ ep, const float* __restrict__ denom,
                         const float* __restrict__ h, float* __restrict__ agg) {
    int e = blockIdx.x;
    int t = threadIdx.x;                 // channel 0..255; head = t>>5
    int s, d;
    if (e < NEDGES) { s = src[e]; d = dst[e]; } else { s = d = e - NEDGES; }
    int head = t >> 5;
    float alpha = ep[(size_t)e * NHEADS + head] / denom[d * NHEADS + head];
    atomicAdd(&agg[(size_t)d * WDIM + t], alpha * h[(size_t)s * WDIM + t]);
}

// ---------- bias + optional ELU (in place OK) ----------
__global__ void bias_act(const float* __restrict__ agg, const float* __restrict__ b,
                         float* __restrict__ out, int apply_elu) {
    long i = (long)blockIdx.x * blockDim.x + threadIdx.x;
    if (i >= (long)NNODES * WDIM) return;
    float v = agg[i] + b[i & (WDIM - 1)];
    if (apply_elu) v = v > 0.f ? v : (expf(v) - 1.0f);
    out[i] = v;
}

// ---------- pooling accumulate ----------
__global__ void pool_accum(const float* __restrict__ x2, const int* __restrict__ batch,
                           unsigned* __restrict__ gmaxenc, float* __restrict__ gsum,
                           float* __restrict__ counts) {
    int n = blockIdx.x, c = threadIdx.x;
    int g = batch[n];
    float v = x2[(size_t)n * WDIM + c];
    atomicMax(&gmaxenc[g * WDIM + c], fenc(v));
    atomicAdd(&gsum[g * WDIM + c], v);
    if (c == 0) atomicAdd(&counts[g], 1.0f);
}

// ---------- pooling finalize: rep = [max | mean] ----------
__global__ void pool_final(const unsigned* __restrict__ gmaxenc, const float* __restrict__ gsum,
                           const float* __restrict__ counts, float* __restrict__ rep) {
    int g = blockIdx.x, c = threadIdx.x;
    float cnt = counts[g];
    float mx = (cnt > 0.f) ? fdec(gmaxenc[g * WDIM + c]) : 0.0f;
    float mean = gsum[g * WDIM + c] / fmaxf(cnt, 1.0f);
    rep[(size_t)g * (2 * WDIM) + c] = mx;
    rep[(size_t)g * (2 * WDIM) + WDIM + c] = mean;
}

// ---------- final FC + sigmoid ----------
__global__ void fc_sigmoid(const float* __restrict__ rep, const float* __restrict__ Wfc,
                           const float* __restrict__ bfc, float* __restrict__ out) {
    __shared__ float red[256];
    int g = blockIdx.x, t = threadIdx.x;
    const float* r = rep + (size_t)g * (2 * WDIM);
    float s = r[t] * Wfc[t] + r[WDIM + t] * Wfc[WDIM + t];
    red[t] = s;
    __syncthreads();
    for (int off = 128; off > 0; off >>= 1) {
        if (t < off) red[t] += red[t + off];
        __syncthreads();
    }
    if (t == 0) out[g] = 1.0f / (1.0f + expf(-(red[0] + bfc[0])));
}

extern "C" void kernel_launch(void* const* d_in, const int* in_sizes, int n_in,
                              void* d_out, int out_size, void* d_ws, size_t ws_size,
                              hipStream_t stream) {
    (void)in_sizes; (void)n_in; (void)out_size; (void)ws_size;

    const float* x    = (const float*)d_in[0];
    const int*   ei   = (const int*)d_in[1];      // [2, NEDGES]: src row, dst row
    const int*   bat  = (const int*)d_in[2];
    const float* W1   = (const float*)d_in[3];
    const float* a1s  = (const float*)d_in[4];
    const float* a1d  = (const float*)d_in[5];
    const float* b1   = (const float*)d_in[6];
    const float* W2   = (const float*)d_in[7];
    const float* a2s  = (const float*)d_in[8];
    const float* a2d  = (const float*)d_in[9];
    const float* b2   = (const float*)d_in[10];
    const float* Wfc  = (const float*)d_in[11];
    const float* bfc  = (const float*)d_in[12];

    float* out = (float*)d_out;        // [256]
    float* rep = out + NGRAPH;         // [256, 512]

    const int* esrc = ei;
    const int* edst = ei + NEDGES;
    const int ETOT = NEDGES + NNODES;

    // workspace carve-up (~137 MB)
    float*    bufA  = (float*)d_ws;                              // h        [N,256]
    float*    bufB  = bufA + (size_t)NNODES * WDIM;              // agg/x    [N,256]
    float*    als   = bufB + (size_t)NNODES * WDIM;              // [N,8]
    float*    ald   = als + (size_t)NNODES * NHEADS;             // [N,8]
    unsigned* menc  = (unsigned*)(ald + (size_t)NNODES * NHEADS);// [N,8]
    float*    denom = (float*)(menc + (size_t)NNODES * NHEADS);  // [N,8]
    float*    ep    = denom + (size_t)NNODES * NHEADS;           // [E+N,8]
    unsigned* gmaxe = (unsigned*)(ep + (size_t)ETOT * NHEADS);   // [256,256]
    float*    gsum  = (float*)(gmaxe + NGRAPH * WDIM);           // [256,256]
    float*    cnts  = gsum + NGRAPH * WDIM;                      // [256]

    const long nNH   = (long)NNODES * NHEADS;
    const long nNW   = (long)NNODES * WDIM;
    const long nEH   = (long)ETOT * NHEADS;
    const int  bNH   = (int)((nNH + 255) / 256);
    const int  bNW   = (int)((nNW + 255) / 256);
    const int  bEH   = (int)((nEH + 255) / 256);

    auto run_layer = [&](const float* xin, int K, const float* Wm, const float* as_,
                         const float* ad_, const float* bias, int elu) {
        // GEMM: xin[N,K] @ Wm[K,256] -> bufA   (one wave per 16x64 strip)
        long waves  = (long)(NNODES / 16) * (WDIM / 64);
        int  blocks = (int)((waves * 32 + 255) / 256);
        gemm_f32_wmma<<<blocks, 256, 0, stream>>>(xin, Wm, bufA, NNODES, WDIM, K);

        compute_al<<<bNH, 256, 0, stream>>>(bufA, as_, ad_, als, ald, NNODES);

        fill_u32<<<bNH, 256, 0, stream>>>(menc, 0u, nNH);
        fill_u32<<<bNH, 256, 0, stream>>>((unsigned*)denom, 0u, nNH);
        fill_u32<<<bNW, 256, 0, stream>>>((unsigned*)bufB, 0u, nNW); // zero agg (after GEMM read)

        edge_logits<<<bEH, 256, 0, stream>>>(esrc, edst, als, ald, ep, menc);
        edge_exp   <<<bEH, 256, 0, stream>>>(edst, ep, menc, denom);
        edge_agg   <<<ETOT, 256, 0, stream>>>(esrc, edst, ep, denom, bufA, bufB);
        bias_act   <<<bNW, 256, 0, stream>>>(bufB, bias, bufB, elu);
    };

    run_layer(x,    128, W1, a1s, a1d, b1, 1);   // layer 1 (+ELU), x1 in bufB
    run_layer(bufB, 256, W2, a2s, a2d, b2, 0);   // layer 2, x2 in bufB

    // pooling
    fill_u32<<<(NGRAPH * WDIM + 255) / 256, 256, 0, stream>>>(gmaxe, 0u, NGRAPH * WDIM);
    fill_u32<<<(NGRAPH * WDIM + 255) / 256, 256, 0, stream>>>((unsigned*)gsum, 0u, NGRAPH * WDIM);
    fill_u32<<<1, 256, 0, stream>>>((unsigned*)cnts, 0u, NGRAPH);
    pool_accum<<<NNODES, 256, 0, stream>>>(bufB, bat, gmaxe, gsum, cnts);
    pool_final<<<NGRAPH, 256, 0, stream>>>(gmaxe, gsum, cnts, rep);
    fc_sigmoid<<<NGRAPH, 256, 0, stream>>>(rep, Wfc, bfc, out);
}